// BoundaryDetectionLoss_20469814133213
// MI455X (gfx1250) — compile-verified
//
#include <hip/hip_runtime.h>

// ---------------- configuration ----------------
#define TPB   256                  // threads per block (8 wave32 waves)
#define EPT   16                   // elements per thread
#define CHUNK (TPB * EPT)          // 4096 elements per block
#define TROW  131072               // row length T (from reference)
#define NEGF  (-1.0e9f)            // reference NEG sentinel
#define BIGF  ( 1.0e9f)

typedef __attribute__((ext_vector_type(2))) float v2f;
typedef __attribute__((ext_vector_type(8))) float v8f;

// ---------------- ds_swizzle xor-shuffle (wave32, immediate offset) ----------------
// group-of-32 mode: offset[15]=0, offset[14:10]=xor_mask, [9:5]=or_mask, [4:0]=and_mask
template <int XOR>
__device__ __forceinline__ float swz_xor(float v) {
  int r = __builtin_amdgcn_ds_swizzle(__float_as_int(v), (XOR << 10) | 0x1f);
  return __int_as_float(r);
}

// ---------------- wave32 reduction via V_WMMA_F32_16X16X4_F32 ----------------
// A = ones(16x4)  =>  D[m,n] = sum_k B[k,n].  Each lane contributes b[0]=v,
// b[1]=0, so the 64 B-slots hold exactly the 32 lane values (+32 zeros).
// All rows of D are identical (A==1), and summing c[0] over one 16-lane half
// equals the sum of ALL B slots = the 32-lane total, independent of the exact
// lane->(k,n) pairing.  4 immediate ds_swizzle xor steps finish it; afterwards
// every lane holds the wave total.
__device__ __forceinline__ float wave_reduce_sum_wmma(float v) {
  v2f a; a[0] = 1.0f; a[1] = 1.0f;
  v2f b; b[0] = v;    b[1] = 0.0f;
  v8f c = {};
  c = __builtin_amdgcn_wmma_f32_16x16x4_f32(false, a, false, b,
                                            (short)0, c, false, false);
  float s = c[0];
  s += swz_xor<8>(s);
  s += swz_xor<4>(s);
  s += swz_xor<2>(s);
  s += swz_xor<1>(s);
  return s;
}

template <int NT>
__device__ __forceinline__ float block_reduce_sum(float v, float* sm, int tid) {
  float ws = wave_reduce_sum_wmma(v);          // full-EXEC, all waves
  int wave = tid >> 5;
  if ((tid & 31) == 0) sm[wave] = ws;
  __syncthreads();
  float tot = 0.0f;
  if (tid == 0) {
#pragma unroll
    for (int i = 0; i < NT / 32; ++i) tot += sm[i];
  }
  __syncthreads();
  return tot;                                   // valid on tid 0
}

__device__ __forceinline__ float wave_reduce_max(float v) {
  v = fmaxf(v, swz_xor<16>(v));
  v = fmaxf(v, swz_xor<8>(v));
  v = fmaxf(v, swz_xor<4>(v));
  v = fmaxf(v, swz_xor<2>(v));
  v = fmaxf(v, swz_xor<1>(v));
  return v;
}
__device__ __forceinline__ float wave_reduce_min(float v) {
  v = fminf(v, swz_xor<16>(v));
  v = fminf(v, swz_xor<8>(v));
  v = fminf(v, swz_xor<4>(v));
  v = fminf(v, swz_xor<2>(v));
  v = fminf(v, swz_xor<1>(v));
  return v;
}

// exclusive prefix-max over the block (threads < tid), seeded with carry
__device__ __forceinline__ float block_scan_excl_max(float v, float seed,
                                                     float* sm, int tid) {
  sm[tid] = v; __syncthreads();
#pragma unroll
  for (int off = 1; off < TPB; off <<= 1) {
    float t = sm[tid];
    if (tid >= off) t = fmaxf(t, sm[tid - off]);
    __syncthreads();
    sm[tid] = t; __syncthreads();
  }
  float r = (tid == 0) ? seed : fmaxf(seed, sm[tid - 1]);
  __syncthreads();
  return r;
}

// exclusive suffix-min over the block (threads > tid), seeded with carry
__device__ __forceinline__ float block_scan_excl_min_rev(float v, float seed,
                                                         float* sm, int tid) {
  int r = TPB - 1 - tid;
  sm[r] = v; __syncthreads();
#pragma unroll
  for (int off = 1; off < TPB; off <<= 1) {
    float t = sm[r];
    if (r >= off) t = fminf(t, sm[r - off]);
    __syncthreads();
    sm[r] = t; __syncthreads();
  }
  float out = (r == 0) ? seed : fminf(seed, sm[r - 1]);
  __syncthreads();
  return out;
}

// ---------------- Pass A: per-chunk boundary position aggregates ----------------
__global__ void __launch_bounds__(TPB)
chunk_agg_kernel(const float* __restrict__ tgt, float* __restrict__ cmax,
                 float* __restrict__ cmin, int nch) {
  int row = blockIdx.y, ch = blockIdx.x, tid = threadIdx.x;
  size_t base = (size_t)row * TROW + (size_t)ch * CHUNK + (size_t)tid * EPT;
  float segpos = (float)(ch * CHUNK + tid * EPT);
  const float4* zp = (const float4*)(tgt + base);
  float tmax = NEGF, tmin = BIGF;
#pragma unroll
  for (int q = 0; q < 4; ++q) {
    float4 z = zp[q];
    float p = segpos + 4.0f * q;
    if (z.x > 0.5f) { tmin = fminf(tmin, p);        tmax = fmaxf(tmax, p); }
    if (z.y > 0.5f) { tmin = fminf(tmin, p + 1.0f); tmax = fmaxf(tmax, p + 1.0f); }
    if (z.z > 0.5f) { tmin = fminf(tmin, p + 2.0f); tmax = fmaxf(tmax, p + 2.0f); }
    if (z.w > 0.5f) { tmin = fminf(tmin, p + 3.0f); tmax = fmaxf(tmax, p + 3.0f); }
  }
  tmax = wave_reduce_max(tmax);
  tmin = wave_reduce_min(tmin);
  __shared__ float smax[TPB / 32], smin[TPB / 32];
  int wave = tid >> 5;
  if ((tid & 31) == 0) { smax[wave] = tmax; smin[wave] = tmin; }
  __syncthreads();
  if (tid == 0) {
    float mx = NEGF, mn = BIGF;
#pragma unroll
    for (int i = 0; i < TPB / 32; ++i) { mx = fmaxf(mx, smax[i]); mn = fminf(mn, smin[i]); }
    cmax[(size_t)row * nch + ch] = mx;
    cmin[(size_t)row * nch + ch] = mn;
  }
}

// ---------------- Pass B: inter-chunk carries (tiny) ----------------
__global__ void carry_kernel(const float* __restrict__ cmax,
                             const float* __restrict__ cmin,
                             float* __restrict__ clast, float* __restrict__ cnext,
                             int nrows, int nch) {
  int row = blockIdx.x * blockDim.x + threadIdx.x;
  if (row >= nrows) return;
  size_t base = (size_t)row * nch;
  float run = NEGF;
  for (int c = 0; c < nch; ++c) { clast[base + c] = run; run = fmaxf(run, cmax[base + c]); }
  run = BIGF;
  for (int c = nch - 1; c >= 0; --c) { cnext[base + c] = run; run = fminf(run, cmin[base + c]); }
}

// ---------------- Pass C: fused distance transform + weighted BCE ----------------
__global__ void __launch_bounds__(TPB)
loss_kernel(const float* __restrict__ logits, const float* __restrict__ tgt,
            const float* __restrict__ clast, const float* __restrict__ cnext,
            float* __restrict__ partials, int nch) {
  __shared__ float sm[TPB];
  int row = blockIdx.y, ch = blockIdx.x, tid = threadIdx.x;
  size_t base = (size_t)row * TROW + (size_t)ch * CHUNK + (size_t)tid * EPT;
  float segpos = (float)(ch * CHUNK + tid * EPT);

  // targets -> 16-bit boundary mask (L2-resident re-read)
  const float4* zp = (const float4*)(tgt + base);
  unsigned mask = 0;
#pragma unroll
  for (int q = 0; q < 4; ++q) {
    float4 z = zp[q];
    mask |= (z.x > 0.5f ? 1u : 0u) << (4 * q + 0);
    mask |= (z.y > 0.5f ? 1u : 0u) << (4 * q + 1);
    mask |= (z.z > 0.5f ? 1u : 0u) << (4 * q + 2);
    mask |= (z.w > 0.5f ? 1u : 0u) << (4 * q + 3);
  }
  float tmax = mask ? segpos + (float)(31 - __clz((int)mask)) : NEGF;
  float tmin = mask ? segpos + (float)(__ffs((int)mask) - 1) : BIGF;

  float seedL = clast[(size_t)row * nch + ch];
  float seedN = cnext[(size_t)row * nch + ch];
  float thLast = block_scan_excl_max(tmax, seedL, sm, tid);      // nearest boundary < my segment
  float thNext = block_scan_excl_min_rev(tmin, seedN, sm, tid);  // nearest boundary > my segment

  float lasts[EPT];
  float run = thLast;
#pragma unroll
  for (int i = 0; i < EPT; ++i) {
    float p = segpos + (float)i;
    if ((mask >> i) & 1u) run = p;
    lasts[i] = run;
  }

  const float4* xp = (const float4*)(logits + base);
  float xf[EPT];
#pragma unroll
  for (int q = 0; q < 4; ++q) {
    float4 x = xp[q];
    xf[4 * q + 0] = x.x; xf[4 * q + 1] = x.y;
    xf[4 * q + 2] = x.z; xf[4 * q + 3] = x.w;
  }

  float acc = 0.0f;
  float nxt = thNext;
#pragma unroll
  for (int i = EPT - 1; i >= 0; --i) {
    float p = segpos + (float)i;
    if ((mask >> i) & 1u) nxt = p;
    float dist = fminf(p - lasts[i], nxt - p);
    float w = 1.0f + expf(-dist * 0.2f);                 // underflows to 1.0 if no boundary
    float zz = ((mask >> i) & 1u) ? 1.0f : 0.0f;
    float x = xf[i];
    float sp = fmaxf(-x, 0.0f) + log1pf(expf(-fabsf(x)));  // stable softplus(-x)
    acc += ((1.0f - zz) * x + (1.0f + zz) * sp) * w;       // pos_weight = 2
  }

  float bsum = block_reduce_sum<TPB>(acc, sm, tid);
  if (tid == 0) partials[(size_t)row * nch + ch] = bsum;
}

// ---------------- final reduction -> (start_loss, end_loss, total) ----------------
__global__ void __launch_bounds__(TPB)
final_kernel(const float* __restrict__ ps, const float* __restrict__ pe,
             float* __restrict__ out, int n, float inv) {
  __shared__ float sm[TPB / 32];
  int tid = threadIdx.x;
  float s = 0.0f, e = 0.0f;
  for (int i = tid; i < n; i += TPB) { s += ps[i]; e += pe[i]; }
  float ss = block_reduce_sum<TPB>(s, sm, tid);
  float ee = block_reduce_sum<TPB>(e, sm, tid);
  if (tid == 0) {
    float sl = ss * inv, el = ee * inv;
    out[0] = sl;
    out[1] = el;
    out[2] = 0.5f * (sl + el);
  }
}

// ---------------- host-side launch ----------------
extern "C" void kernel_launch(void* const* d_in, const int* in_sizes, int n_in,
                              void* d_out, int out_size, void* d_ws, size_t ws_size,
                              hipStream_t stream) {
  const float* start_probs   = (const float*)d_in[0];
  const float* end_probs     = (const float*)d_in[1];
  const float* start_targets = (const float*)d_in[2];
  const float* end_targets   = (const float*)d_in[3];

  int N   = in_sizes[0];
  int T   = TROW;
  int Bn  = N / T;                 // 64
  int nch = T / CHUNK;             // 32 chunks per row
  int totalChunks = Bn * nch;      // 2048

  float* ws     = (float*)d_ws;
  float* cmaxS  = ws;
  float* cminS  = cmaxS  + totalChunks;
  float* cmaxE  = cminS  + totalChunks;
  float* cminE  = cmaxE  + totalChunks;
  float* clastS = cminE  + totalChunks;
  float* cnextS = clastS + totalChunks;
  float* clastE = cnextS + totalChunks;
  float* cnextE = clastE + totalChunks;
  float* partS  = cnextE + totalChunks;
  float* partE  = partS  + totalChunks;

  dim3 grid(nch, Bn), block(TPB);

  chunk_agg_kernel<<<grid, block, 0, stream>>>(start_targets, cmaxS, cminS, nch);
  chunk_agg_kernel<<<grid, block, 0, stream>>>(end_targets,   cmaxE, cminE, nch);

  int cb = (Bn + 63) / 64;
  carry_kernel<<<cb, 64, 0, stream>>>(cmaxS, cminS, clastS, cnextS, Bn, nch);
  carry_kernel<<<cb, 64, 0, stream>>>(cmaxE, cminE, clastE, cnextE, Bn, nch);

  loss_kernel<<<grid, block, 0, stream>>>(start_probs, start_targets, clastS, cnextS, partS, nch);
  loss_kernel<<<grid, block, 0, stream>>>(end_probs,   end_targets,   clastE, cnextE, partE, nch);

  float inv = 1.0f / ((float)Bn * (float)T);
  final_kernel<<<1, TPB, 0, stream>>>(partS, partE, (float*)d_out, totalChunks, inv);
}